// BayesianLoss_22763326668833
// MI455X (gfx1250) — compile-verified
//
#include <hip/hip_runtime.h>
#include <hip/hip_bf16.h>

// Problem geometry (fixed by the reference): logits [1,21,1024,1024] f32,
// masks [1,1,1024,1024] i32, out [1,1024] f32.
#define CCH   21
#define WW    1024
#define HH    1024
#define HWSZ  (WW * HH)            // 1<<20
#define TPB   256
#define ROWS  16                   // rows per block
#define CHUNKS (HH / ROWS)         // 64
#define NBLK  ((WW / TPB) * CHUNKS) // 256 blocks in kernel 1

typedef __attribute__((address_space(1))) int        gint_t;   // global AS
typedef __attribute__((address_space(3))) int        lint_t;   // LDS AS
typedef __attribute__((address_space(3))) float      lfloat_t;

#if __has_builtin(__builtin_amdgcn_global_load_async_to_lds_b32)
#define HAVE_ASYNC_BUILTIN 1
#else
#define HAVE_ASYNC_BUILTIN 0
#endif

#if __has_builtin(__builtin_amdgcn_s_wait_asynccnt)
#define WAIT_ASYNC(n) do { __builtin_amdgcn_s_wait_asynccnt(n); \
                           asm volatile("" ::: "memory"); } while (0)
#else
#define WAIT_ASYNC(n) asm volatile("s_wait_asynccnt %0" :: "i"(n) : "memory")
#endif

// Issue 21 async global->LDS loads for one row tile: element (c, h, w=base+tid)
// lands at tile[c][tid]. Each lane later reads back only its own column, so no
// cross-thread barrier is required -- only ASYNCcnt completion.
__device__ __forceinline__ void issue_row(const float* __restrict__ logits,
                                          float (*tile)[TPB],
                                          int h, int w, int tid) {
#pragma unroll
  for (int c = 0; c < CCH; ++c) {
    const float* g = logits + (size_t)c * HWSZ + (size_t)h * WW + w;
#if HAVE_ASYNC_BUILTIN
    __builtin_amdgcn_global_load_async_to_lds_b32(
        (gint_t*)(float*)g, (lint_t*)&tile[c][tid], 0, 0);
#else
    int voff = (c * HWSZ + h * WW + w) * 4;  // < 88MB, fits i32
    asm volatile("global_load_async_to_lds_b32 %0, %1, %2"
                 :: "v"((lfloat_t*)&tile[c][tid]), "v"(voff), "s"(logits)
                 : "memory");
#endif
  }
}

__global__ __launch_bounds__(TPB) void bayes_loss_main(
    const float* __restrict__ logits, const int* __restrict__ masks,
    float* __restrict__ epiP, float* __restrict__ aleaP) {
  __shared__ float tile[2][CCH][TPB];   // 43,008 B double buffer
  __shared__ float red[TPB];

  const int tid = threadIdx.x;
  const int w   = blockIdx.x * TPB + tid;
  const int h0  = blockIdx.y * ROWS;

  float epi_acc  = 0.f;
  float alea_acc = 0.f;

  issue_row(logits, tile[0], h0, w, tid);          // prime the pipeline

  for (int r = 0; r < ROWS; ++r) {
    const int h = h0 + r;
    if (r + 1 < ROWS) {
      // Buffer (r+1)&1 was last read at iteration r-1: make sure those
      // ds_loads retired before the async engine overwrites it.
      asm volatile("s_wait_dscnt 0" ::: "memory");
      issue_row(logits, tile[(r + 1) & 1], h + 1, w, tid);
      WAIT_ASYNC(CCH);   // <=21 outstanding => row r's 21 loads complete
    } else {
      WAIT_ASYNC(0);
    }

    float v[CCH];
#pragma unroll
    for (int c = 0; c < CCH; ++c) v[c] = tile[r & 1][c][tid];

    float m = v[0];
#pragma unroll
    for (int c = 1; c < CCH; ++c) m = fmaxf(m, v[c]);

    float s = 0.f;
#pragma unroll
    for (int c = 0; c < CCH; ++c) s += __expf(v[c] - m);
    const float inv_s = 1.0f / s;
    const float logs  = __logf(s);

    float alea = 0.f, epi = 0.f;
#pragma unroll
    for (int c = 0; c < CCH; ++c) {
      const float x  = v[c] - m;
      const float p  = __expf(x) * inv_s;        // softmax
      const float lp = __logf(p + 1e-10f);       // log(softmax + eps)
      const float ls = x - logs;                 // log_softmax
      alea -= lp;                                // sum_c(-lp)
      // xlogy(ls, ls): 0 when ls==0, else ls*log(ls) (NaN for ls<0, faithful)
      const float xl = (ls == 0.f) ? 0.f : ls * __logf(ls);
      epi += xl - ls * lp;
    }

    epi_acc += epi;
    const int mk = masks[(size_t)h * WW + w];
    alea_acc += (mk != 255) ? alea : 0.f;
  }

  // Deterministic per-(chunk, w) epistemic partial.
  epiP[(size_t)blockIdx.y * WW + w] = epi_acc;

  // Deterministic block tree-reduction of the aleatoric partial.
  red[tid] = alea_acc;
  __syncthreads();
#pragma unroll
  for (int off = TPB / 2; off > 0; off >>= 1) {
    if (tid < off) red[tid] += red[tid + off];
    __syncthreads();
  }
  if (tid == 0) aleaP[blockIdx.y * gridDim.x + blockIdx.x] = red[0];
}

__global__ __launch_bounds__(256) void bayes_loss_finalize(
    const float* __restrict__ epiP, const float* __restrict__ aleaP,
    float* __restrict__ out) {
  const int w = blockIdx.x * blockDim.x + threadIdx.x;
  float e = 0.f;
  for (int k = 0; k < CHUNKS; ++k) e += epiP[(size_t)k * WW + w];
  float a = 0.f;
  for (int i = 0; i < NBLK; ++i) a += aleaP[i];   // redundant but deterministic
  out[w] = a * (1.0f / (float)HWSZ) + e * (1.0f / (float)HH);
}

extern "C" void kernel_launch(void* const* d_in, const int* in_sizes, int n_in,
                              void* d_out, int out_size, void* d_ws, size_t ws_size,
                              hipStream_t stream) {
  (void)in_sizes; (void)n_in; (void)out_size; (void)ws_size;
  const float* logits = (const float*)d_in[0];
  const int*   masks  = (const int*)d_in[1];
  float* out  = (float*)d_out;
  float* epiP = (float*)d_ws;                       // 64*1024 floats
  float* aleaP = epiP + (size_t)CHUNKS * WW;        // 256 floats

  dim3 grid1(WW / TPB, CHUNKS);                     // 4 x 64 = 256 blocks
  bayes_loss_main<<<grid1, TPB, 0, stream>>>(logits, masks, epiP, aleaP);
  bayes_loss_finalize<<<WW / 256, 256, 0, stream>>>(epiP, aleaP, out);
}